// DecoderRNN_12309376270952
// MI455X (gfx1250) — compile-verified
//
#include <hip/hip_runtime.h>
#include <hip/hip_bf16.h>
#include <math.h>

#define VOC 32000
#define HID 512
#define BAT 32
#define SEQ 64
#define MROWS (BAT * SEQ)   // 2048
#define KIN   (2 * HID)     // 1024
#define NG3   (3 * HID)     // 1536
#define APAD  516           // 512 + 4-float pad -> conflict-free ds_read_b64

typedef float v2f __attribute__((ext_vector_type(2)));
typedef float v8f __attribute__((ext_vector_type(8)));

__device__ __forceinline__ v8f wmma_f32_4(v2f a, v2f b, v8f c) {
    // D = A(16x4) * B(4x16) + C(16x16), fp32, wave32
    return __builtin_amdgcn_wmma_f32_16x16x4_f32(false, a, false, b, (short)0, c, false, false);
}

__device__ __forceinline__ v8f splat8(float v) {
    v8f r;
#pragma unroll
    for (int i = 0; i < 8; ++i) r[i] = v;
    return r;
}

// ---------------------------------------------------------------------------
// Kernel 1: xp[m,n] = relu(concat(emb[tok[m]], thought[b])) @ W_ih^T + b_ih
// M=2048, K=1024, N=1536.  One wave per 16x64 slab: A fragment reused 4x.
// ---------------------------------------------------------------------------
__global__ __launch_bounds__(256) void k_inproj(
    const int* __restrict__ seqs, const float* __restrict__ thought,
    const float* __restrict__ emb, const float* __restrict__ W_ih,
    const float* __restrict__ b_ih, float* __restrict__ xp)
{
    const int lane  = threadIdx.x & 31;
    const int wave  = blockIdx.x * (blockDim.x >> 5) + (threadIdx.x >> 5);
    const int mt    = wave / 24;        // 128 M-tiles
    const int ng    = wave % 24;        // 24 groups of 64 cols
    const int mrow  = mt * 16 + (lane & 15);
    const int khalf = lane >> 4;        // selects k-pair {0,1} or {2,3}
    const int ncol  = lane & 15;
    const int nbase = ng * 64;
    const int b     = mrow >> 6;        // mrow / SEQ
    const int tok   = seqs[mrow];
    const float* arow_e = emb + (size_t)tok * HID;   // k in [0,512)
    const float* arow_t = thought + (size_t)b * HID; // k-512 in [0,512)

    v8f acc[4];
#pragma unroll
    for (int t = 0; t < 4; ++t) acc[t] = splat8(b_ih[nbase + t * 16 + ncol]);

    for (int kb = 0; kb < KIN; kb += 4) {
        const int k0 = kb + 2 * khalf;                 // uniform side of 512
        float2 av = (k0 < HID) ? *(const float2*)(arow_e + k0)
                               : *(const float2*)(arow_t + (k0 - HID));
        v2f a; a.x = fmaxf(av.x, 0.f); a.y = fmaxf(av.y, 0.f);
#pragma unroll
        for (int t = 0; t < 4; ++t) {
            const int n = nbase + t * 16 + ncol;
            float2 bv = *(const float2*)(W_ih + (size_t)n * KIN + k0);
            v2f bf; bf.x = bv.x; bf.y = bv.y;
            acc[t] = wmma_f32_4(a, bf, acc[t]);
        }
    }
    const int rbase = 8 * khalf;       // C/D layout: lanes16-31 hold M=8..15
#pragma unroll
    for (int t = 0; t < 4; ++t) {
        const int n = nbase + t * 16 + ncol;
#pragma unroll
        for (int i = 0; i < 8; ++i)
            xp[(size_t)(mt * 16 + rbase + i) * NG3 + n] = acc[t][i];
    }
}

// ---------------------------------------------------------------------------
// Kernel 2: persistent single-workgroup GRU scan. h (64KB) and hp (192KB)
// live in LDS (256KB dynamic). 32 waves; 192 hp-tiles => 6 tiles/wave.
// ---------------------------------------------------------------------------
__global__ __launch_bounds__(1024) void k_gru(
    const float* __restrict__ xp, const float* __restrict__ W_hh,
    const float* __restrict__ b_hh, float* __restrict__ ys,
    float* __restrict__ hT)
{
    extern __shared__ float lds[];
    float* h  = lds;                    // [BAT][HID]   16384 f
    float* hp = lds + BAT * HID;        // [BAT][NG3]   49152 f
    const int tid   = threadIdx.x;
    const int lane  = tid & 31;
    const int wave  = tid >> 5;         // 0..31
    const int khalf = lane >> 4;

    for (int e = tid; e < BAT * HID; e += 1024) h[e] = 0.f;
    __syncthreads();

    for (int s = 0; s < SEQ; ++s) {
        // hp = h @ W_hh^T + b_hh   (M=32, K=512, N=1536)
        for (int t6 = 0; t6 < 6; ++t6) {
            const int t    = wave * 6 + t6;     // 0..191
            const int mt   = t & 1;
            const int nt   = t >> 1;
            const int mrow = mt * 16 + (lane & 15);
            const int ncol = nt * 16 + (lane & 15);
            v8f acc = splat8(b_hh[ncol]);
            for (int kb = 0; kb < HID; kb += 4) {
                const int k0 = kb + 2 * khalf;
                float2 av = *(const float2*)(h + mrow * HID + k0);
                float2 bv = *(const float2*)(W_hh + (size_t)ncol * HID + k0);
                v2f a; a.x = av.x; a.y = av.y;
                v2f bf; bf.x = bv.x; bf.y = bv.y;
                acc = wmma_f32_4(a, bf, acc);
            }
            const int rbase = 8 * khalf;
#pragma unroll
            for (int i = 0; i < 8; ++i)
                hp[(mt * 16 + rbase + i) * NG3 + ncol] = acc[i];
        }
        __syncthreads();
        // gate math + emit ys[:,s,:]
        for (int e = tid; e < BAT * HID; e += 1024) {
            const int bb = e >> 9;      // /512
            const int j  = e & 511;
            const float* xrow = xp + (size_t)(bb * SEQ + s) * NG3;
            const float xr = xrow[j], xz = xrow[HID + j], xn = xrow[2 * HID + j];
            const float hr = hp[bb * NG3 + j];
            const float hz = hp[bb * NG3 + HID + j];
            const float hn = hp[bb * NG3 + 2 * HID + j];
            const float hprev = h[e];
            const float r = 1.f / (1.f + __expf(-(xr + hr)));
            const float z = 1.f / (1.f + __expf(-(xz + hz)));
            const float n = tanhf(xn + r * hn);
            const float hnew = (1.f - z) * n + z * hprev;
            ys[(size_t)(bb * SEQ + s) * HID + j] = hnew;
            h[e] = hnew;
        }
        __syncthreads();
    }
    for (int e = tid; e < BAT * HID; e += 1024) hT[e] = h[e];
}

// ---------------------------------------------------------------------------
// Kernel 3: logits = ys @ W_out^T + b_out   (M=2048, K=512, N=32000)
// Block = 8 waves sharing a 32x512 A slab in LDS (async-copied, padded to
// stride 516 for conflict-free ds_read_b64). Each wave: 16x64 output slab.
// Grid: (2048/32) x (32000/256) = 64 x 125 = 8000 blocks, exact.
// ---------------------------------------------------------------------------
__global__ __launch_bounds__(256) void k_logits(
    const float* __restrict__ ys, const float* __restrict__ W_out,
    const float* __restrict__ b_out, float* __restrict__ logits)
{
    extern __shared__ float As[];       // [32][APAD] = 66048 bytes
    const int tid    = threadIdx.x;
    const int lane   = tid & 31;
    const int wavein = tid >> 5;        // 0..7
    const int bm     = blockIdx.x / 125;    // 64 row-blocks of 32
    const int bn     = blockIdx.x % 125;    // 125 col-blocks of 256

    // --- async copy A slab: ys[bm*32 .. +32, 0..512) -> LDS (b128 lanes) ---
    {
        const unsigned asbase = (unsigned)(uintptr_t)As;  // LDS offset (addr[31:0])
#pragma unroll
        for (int j = 0; j < 16; ++j) {
            const int i = tid + j * 256;       // 4096 x float4 transfers
            const int e = i * 4;
            const int r = e >> 9;              // /512
            const int c = e & 511;
            const unsigned loff = asbase + (unsigned)(r * APAD + c) * 4u;
            const float* gp = ys + (size_t)(bm * 32 + r) * HID + c;
            asm volatile("global_load_async_to_lds_b128 %0, %1, off"
                         :: "v"(loff), "v"(gp) : "memory");
        }
        asm volatile("s_wait_asynccnt 0x0" ::: "memory");
    }
    __syncthreads();

    const int mt_l   = wavein & 1;      // local M tile (0/1)
    const int ngl    = wavein >> 1;     // local N group (0..3), 64 cols each
    const int khalf  = lane >> 4;
    const int ncol   = lane & 15;
    const int nbase  = bn * 256 + ngl * 64;
    const float* arow = As + (mt_l * 16 + (lane & 15)) * APAD;

    v8f acc[4];
#pragma unroll
    for (int t = 0; t < 4; ++t) acc[t] = splat8(b_out[nbase + t * 16 + ncol]);

    for (int kb = 0; kb < HID; kb += 4) {
        if ((kb & 127) == 0 && kb + 128 < HID) {
#pragma unroll
            for (int t = 0; t < 4; ++t)
                __builtin_prefetch(W_out + (size_t)(nbase + t * 16 + ncol) * HID
                                         + kb + 128, 0, 1);
        }
        const int k0 = kb + 2 * khalf;
        float2 av = *(const float2*)(arow + k0);   // LDS, conflict-free
        v2f a; a.x = av.x; a.y = av.y;
#pragma unroll
        for (int t = 0; t < 4; ++t) {
            const int n = nbase + t * 16 + ncol;
            float2 bv = *(const float2*)(W_out + (size_t)n * HID + k0);
            v2f bf; bf.x = bv.x; bf.y = bv.y;
            acc[t] = wmma_f32_4(a, bf, acc[t]);
        }
    }
    const int rbase = 8 * khalf;
#pragma unroll
    for (int t = 0; t < 4; ++t) {
        const int n = nbase + t * 16 + ncol;
#pragma unroll
        for (int i = 0; i < 8; ++i)
            logits[(size_t)(bm * 32 + mt_l * 16 + rbase + i) * VOC + n] = acc[t][i];
    }
}

// ---------------------------------------------------------------------------
// Kernel 4: in-place log_softmax over V per row (block per row).
// ---------------------------------------------------------------------------
__global__ __launch_bounds__(256) void k_logsoftmax(float* __restrict__ logp)
{
    float* x = logp + (size_t)blockIdx.x * VOC;
    __shared__ float red[256];
    const int tid = threadIdx.x;

    float mx = -INFINITY;
    for (int i = tid; i < VOC; i += 256) mx = fmaxf(mx, x[i]);
    red[tid] = mx; __syncthreads();
    for (int o = 128; o > 0; o >>= 1) {
        if (tid < o) red[tid] = fmaxf(red[tid], red[tid + o]);
        __syncthreads();
    }
    mx = red[0]; __syncthreads();

    float sum = 0.f;
    for (int i = tid; i < VOC; i += 256) sum += __expf(x[i] - mx);
    red[tid] = sum; __syncthreads();
    for (int o = 128; o > 0; o >>= 1) {
        if (tid < o) red[tid] += red[tid + o];
        __syncthreads();
    }
    const float lse = mx + __logf(red[0]);
    __syncthreads();

    for (int i = tid; i < VOC; i += 256) x[i] -= lse;
}

// ---------------------------------------------------------------------------
extern "C" void kernel_launch(void* const* d_in, const int* in_sizes, int n_in,
                              void* d_out, int out_size, void* d_ws, size_t ws_size,
                              hipStream_t stream)
{
    (void)in_sizes; (void)n_in; (void)out_size; (void)ws_size;
    const int*   seqs    = (const int*)d_in[0];
    const float* thought = (const float*)d_in[1];
    const float* emb     = (const float*)d_in[2];
    const float* W_ih    = (const float*)d_in[3];
    const float* W_hh    = (const float*)d_in[4];
    const float* b_ih    = (const float*)d_in[5];
    const float* b_hh    = (const float*)d_in[6];
    const float* W_out   = (const float*)d_in[7];
    const float* b_out   = (const float*)d_in[8];

    float* logp = (float*)d_out;                       // [2048, 32000]
    float* hT   = logp + (size_t)MROWS * VOC;          // [1, 32, 512]
    float* ws   = (float*)d_ws;
    float* xp   = ws;                                  // [2048, 1536]
    float* ysb  = ws + (size_t)MROWS * NG3;            // [2048, 512]

    // 1) input projection: 128*24 = 3072 waves -> 384 blocks of 8 waves
    hipLaunchKernelGGL(k_inproj, dim3(384), dim3(256), 0, stream,
                       seqs, thought, emb, W_ih, b_ih, xp);
    // 2) GRU scan: single persistent workgroup, 256KB LDS
    hipLaunchKernelGGL(k_gru, dim3(1), dim3(1024),
                       (size_t)(BAT * HID + BAT * NG3) * sizeof(float), stream,
                       xp, W_hh, b_hh, ysb, hT);
    // 3) logits: 8000 blocks x 8 waves, 66KB LDS A-slab per block
    hipLaunchKernelGGL(k_logits, dim3(8000), dim3(256),
                       (size_t)(32 * APAD) * sizeof(float), stream,
                       ysb, W_out, b_out, logp);
    // 4) log_softmax, one block per row
    hipLaunchKernelGGL(k_logsoftmax, dim3(MROWS), dim3(256), 0, stream, logp);
}